// LatentGridConstructor_48172353192555
// MI455X (gfx1250) — compile-verified
//
#include <hip/hip_runtime.h>

typedef float v2f __attribute__((ext_vector_type(2)));
typedef float v4f __attribute__((ext_vector_type(4)));
typedef float v8f __attribute__((ext_vector_type(8)));

#define BB 4
#define EE 256
#define DD 128
#define HH 1024
#define GG 512

// Gaussian kernel: exp(-{1,0,1}/2) normalized
#define GK0 0.2740686f
#define GK1 0.4518628f
#define GK2 0.2740686f

// ---------------------------------------------------------------------------
// WMMA fp32 GEMM: C[m,n] = sum_k A[m,k]*W[n,k] (+bias[n]) (*rowmask[m])
// A: M x K row-major (lda), W: N x K row-major (ldw; allows W_fus column slice)
//
// Register blocking: each wave owns a 2x4 block of 16x16 tiles (32 rows x 64
// cols), 8 v8f accumulators. Per K=4 step: 2 A-loads + 4 B-loads -> 8
// v_wmma_f32_16x16x4_f32 (~10.7 FLOP/byte from L2, 2.7x the single-tile version).
//
// fp32 A-16x4 layout: lane l holds K = 2*(l/16)+{0,1} for row M = l%16
// fp32 B-4x16 layout: lane l holds K = 2*(l/16)+{0,1} for col N = l%16
//   -> each fragment is one float2 load at (row)*ld + k + 2*(l>>4)
// C/D layout: VGPR v, lane l -> row = 8*(l/16)+v, col = l%16
// ---------------------------------------------------------------------------
__global__ __launch_bounds__(256) void wmma_gemm_f32(
    const float* __restrict__ A, int lda,
    const float* __restrict__ W, int ldw,
    const float* __restrict__ bias,
    const float* __restrict__ rowmask,
    float* __restrict__ C, int ldc,
    int M, int N, int K)
{
    const int wave    = threadIdx.x >> 5;
    const int lane    = threadIdx.x & 31;
    const int nBlk    = N >> 6;                    // 64-wide col blocks
    const int waveId  = blockIdx.x * 8 + wave;
    const int mb      = waveId / nBlk;             // 32-row block index
    const int nb      = waveId % nBlk;             // 64-col block index
    if (mb * 32 >= M) return;                      // wave-uniform guard

    const int r    = lane & 15;
    const int koff = (lane >> 4) << 1;             // 0 or 2

    const float* __restrict__ Ap0 = A + (size_t)(mb * 32 + r)      * lda + koff;
    const float* __restrict__ Ap1 = A + (size_t)(mb * 32 + 16 + r) * lda + koff;
    const float* __restrict__ Wp0 = W + (size_t)(nb * 64 + r)      * ldw + koff;
    const float* __restrict__ Wp1 = Wp0 + (size_t)16 * ldw;
    const float* __restrict__ Wp2 = Wp0 + (size_t)32 * ldw;
    const float* __restrict__ Wp3 = Wp0 + (size_t)48 * ldw;

    v8f acc00 = {}, acc01 = {}, acc02 = {}, acc03 = {};
    v8f acc10 = {}, acc11 = {}, acc12 = {}, acc13 = {};

#pragma unroll 2
    for (int k = 0; k < K; k += 4) {
        v2f a0 = *(const v2f*)(Ap0 + k);
        v2f a1 = *(const v2f*)(Ap1 + k);
        v2f b0 = *(const v2f*)(Wp0 + k);
        v2f b1 = *(const v2f*)(Wp1 + k);
        v2f b2 = *(const v2f*)(Wp2 + k);
        v2f b3 = *(const v2f*)(Wp3 + k);
        acc00 = __builtin_amdgcn_wmma_f32_16x16x4_f32(false, a0, false, b0, (short)0, acc00, false, false);
        acc01 = __builtin_amdgcn_wmma_f32_16x16x4_f32(false, a0, false, b1, (short)0, acc01, false, false);
        acc02 = __builtin_amdgcn_wmma_f32_16x16x4_f32(false, a0, false, b2, (short)0, acc02, false, false);
        acc03 = __builtin_amdgcn_wmma_f32_16x16x4_f32(false, a0, false, b3, (short)0, acc03, false, false);
        acc10 = __builtin_amdgcn_wmma_f32_16x16x4_f32(false, a1, false, b0, (short)0, acc10, false, false);
        acc11 = __builtin_amdgcn_wmma_f32_16x16x4_f32(false, a1, false, b1, (short)0, acc11, false, false);
        acc12 = __builtin_amdgcn_wmma_f32_16x16x4_f32(false, a1, false, b2, (short)0, acc12, false, false);
        acc13 = __builtin_amdgcn_wmma_f32_16x16x4_f32(false, a1, false, b3, (short)0, acc13, false, false);
    }

    v8f accs[2][4] = {{acc00, acc01, acc02, acc03}, {acc10, acc11, acc12, acc13}};

#pragma unroll
    for (int j = 0; j < 4; ++j) {
        const int   col = nb * 64 + j * 16 + r;
        const float bv  = bias ? bias[col] : 0.0f;
#pragma unroll
        for (int i = 0; i < 2; ++i) {
            const int rowbase = mb * 32 + i * 16 + ((lane >> 4) << 3);
#pragma unroll
            for (int v = 0; v < 8; ++v) {
                const int row = rowbase + v;
                float val = accs[i][j][v] + bv;
                if (rowmask) val *= rowmask[row];
                C[(size_t)row * ldc + col] = val;
            }
        }
    }
}

// ---------------------------------------------------------------------------
// grid[b,d,e,o] = dec_term[b,d,o] + enc_term[b,e,o] + b_fus[o]
// 256 MB streamed write -> non-temporal stores (keep L2 for the intermediates)
// One block per (b,d,e) row of G=512; 128 threads x float4.
// ---------------------------------------------------------------------------
__global__ __launch_bounds__(128) void grid_assemble(
    const float* __restrict__ dec_term,   // [B*D, G]
    const float* __restrict__ enc_term,   // [B*E, G]
    const float* __restrict__ b_fus,      // [G]
    float* __restrict__ grid_out)         // [B, D, E, G]
{
    const int rowIdx = blockIdx.x;        // ((b*D + d)*E + e)
    const int e  = rowIdx & (EE - 1);
    const int bd = rowIdx >> 8;           // b*D + d
    const int b  = bd >> 7;               // D = 128
    const int be = b * EE + e;

    const int o = threadIdx.x << 2;       // 128 threads * 4 = 512
    v4f dv = *(const v4f*)(dec_term + (size_t)bd * GG + o);
    v4f ev = *(const v4f*)(enc_term + (size_t)be * GG + o);
    v4f bf = *(const v4f*)(b_fus + o);
    v4f out = dv + ev + bf;
    __builtin_nontemporal_store(out, (v4f*)(grid_out + (size_t)rowIdx * GG + o));
}

// ---------------------------------------------------------------------------
// mask[b,d,e] = || blur_d(blur_e(grid)) ||_o > 0
// Separability: blurred = w_e(e)*blurD(dec_term)[b,d,:] +
//                         w_d(d)*blurE(enc_term)[b,e,:] +
//                         w_e(e)*w_d(d)*b_fus[:]
// (zero-padded 3-tap blur; w = sum of in-range kernel taps at the edge)
// One block per (b,d): stage blurD row + b_fus in LDS; thread e reduces over o.
// Never re-reads the 256 MB grid -> saves ~768 MB of HBM traffic vs naive.
// ---------------------------------------------------------------------------
__global__ __launch_bounds__(256) void mask_kernel(
    const float* __restrict__ dec_term,   // [B*D, G]
    const float* __restrict__ enc_term,   // [B*E, G]
    const float* __restrict__ b_fus,
    float* __restrict__ mask_out)         // [B, D, E]
{
    __shared__ float sAd[GG];
    __shared__ float sC[GG];

    const int bd = blockIdx.x;            // b*D + d
    const int b  = bd >> 7;
    const int d  = bd & (DD - 1);

    for (int o = threadIdx.x; o < GG; o += blockDim.x) {
        float v = GK1 * dec_term[(size_t)bd * GG + o];
        if (d > 0)      v += GK0 * dec_term[(size_t)(bd - 1) * GG + o];
        if (d < DD - 1) v += GK2 * dec_term[(size_t)(bd + 1) * GG + o];
        sAd[o] = v;
        sC[o]  = b_fus[o];
    }
    __syncthreads();

    const int e = threadIdx.x;            // blockDim == E == 256
    float we = 1.0f;
    if (e == 0)           we = GK1 + GK2;
    else if (e == EE - 1) we = GK0 + GK1;
    float wd = 1.0f;
    if (d == 0)           wd = GK1 + GK2;
    else if (d == DD - 1) wd = GK0 + GK1;

    const float* __restrict__ Br = enc_term + (size_t)(b * EE + e) * GG;
    const bool hasM = (e > 0), hasP = (e < EE - 1);
    const float wew = we * wd;

    float acc = 0.0f;
    for (int o = 0; o < GG; ++o) {
        float bev = GK1 * Br[o];
        if (hasM) bev += GK0 * Br[o - GG];
        if (hasP) bev += GK2 * Br[o + GG];
        float val = we * sAd[o] + wd * bev + wew * sC[o];
        acc += val * val;
    }
    mask_out[(size_t)bd * EE + e] = (acc > 0.0f) ? 1.0f : 0.0f;
}

extern "C" void kernel_launch(void* const* d_in, const int* in_sizes, int n_in,
                              void* d_out, int out_size, void* d_ws, size_t ws_size,
                              hipStream_t stream) {
    const float* enc_h  = (const float*)d_in[0];  // (4,256,1024)
    const float* dec_h  = (const float*)d_in[1];  // (4,128,1024)
    const float* enc_m  = (const float*)d_in[2];  // (4,256)
    const float* dec_m  = (const float*)d_in[3];  // (4,128)
    const float* W_enc  = (const float*)d_in[4];  // (512,1024)
    const float* b_enc  = (const float*)d_in[5];  // (512)
    const float* W_dec  = (const float*)d_in[6];  // (512,1024)
    const float* b_dec  = (const float*)d_in[7];  // (512)
    const float* W_fus  = (const float*)d_in[8];  // (512,1024)
    const float* b_fus  = (const float*)d_in[9];  // (512)

    float* grid_out = (float*)d_out;                                   // 4*128*256*512
    float* mask_out = grid_out + (size_t)BB * DD * EE * GG;            // 4*128*256

    float* ws       = (float*)d_ws;
    float* enc_proj = ws;                         // [1024, 512]  2 MB
    float* dec_proj = enc_proj + 1024 * GG;       // [ 512, 512]  1 MB
    float* enc_term = dec_proj +  512 * GG;       // [1024, 512]  2 MB
    float* dec_term = enc_term + 1024 * GG;       // [ 512, 512]  1 MB

    // Waves per GEMM: (M/32) * (N/64); 8 waves per 256-thread block.
    // 1) enc_proj = (enc_h @ W_enc^T + b_enc) * enc_mask   M=1024 N=512 K=1024
    wmma_gemm_f32<<<(32 * 8) / 8, 256, 0, stream>>>(
        enc_h, HH, W_enc, HH, b_enc, enc_m, enc_proj, GG, BB * EE, GG, HH);

    // 2) dec_proj = (dec_h @ W_dec^T + b_dec) * dec_mask   M=512 N=512 K=1024
    wmma_gemm_f32<<<(16 * 8) / 8, 256, 0, stream>>>(
        dec_h, HH, W_dec, HH, b_dec, dec_m, dec_proj, GG, BB * DD, GG, HH);

    // 3) enc_term = enc_proj @ Wf_e^T  (Wf_e = W_fus[:, :512], ldw=1024)
    wmma_gemm_f32<<<(32 * 8) / 8, 256, 0, stream>>>(
        enc_proj, GG, W_fus, 2 * GG, nullptr, nullptr, enc_term, GG, BB * EE, GG, GG);

    // 4) dec_term = dec_proj @ Wf_d^T  (Wf_d = W_fus[:, 512:], ldw=1024)
    wmma_gemm_f32<<<(16 * 8) / 8, 256, 0, stream>>>(
        dec_proj, GG, W_fus + GG, 2 * GG, nullptr, nullptr, dec_term, GG, BB * DD, GG, GG);

    // 5) grid = dec_term[:, :, None, :] + enc_term[:, None, :, :] + b_fus  (256 MB, NT stores)
    grid_assemble<<<BB * DD * EE, 128, 0, stream>>>(dec_term, enc_term, b_fus, grid_out);

    // 6) mask from separable blur identity (never re-reads the 256 MB grid)
    mask_kernel<<<BB * DD, 256, 0, stream>>>(dec_term, enc_term, b_fus, mask_out);
}